// PrecisionRecall_30477087932512
// MI455X (gfx1250) — compile-verified
//
#include <hip/hip_runtime.h>

typedef __attribute__((ext_vector_type(2))) float v2f;
typedef __attribute__((ext_vector_type(8))) float v8f;

#define DFEAT 2048
#define BM 128          // rows per workgroup (8 waves x 16)
#define BN 128          // cols per chunk (8 WMMA tiles)
#define KC 16           // K-chunk staged in LDS (double-buffered)
#define LDA 20          // padded LDS stride (floats): 80 B rows, 16B aligned, conflict-free
#define SCR_LD 66       // padded scratch stride for scan phase
#define NN 8            // capacity of per-row smallest list (supports k <= 7)
#define GY 8            // column-chunk grid split
#define MAXN 10016

#define WAIT_ASYNC0() asm volatile("s_wait_asynccnt 0x0" ::: "memory")

// Async DMA: global -> LDS, 16B per lane, tracked by ASYNCcnt (no VGPR data path).
__device__ __forceinline__ void stage_async(const float* __restrict__ X, int base, int N,
                                            int kc, float* lds, int tid) {
#pragma unroll
  for (int j = 0; j < (BM * (KC / 4)) / 256; ++j) {  // 2 chunks of 16B per thread
    int i = tid + j * 256;
    int row = i >> 2, c4 = (i & 3) << 2;
    int gr = base + row;
    if (gr >= N) gr = N - 1;
    const float* src = X + (size_t)gr * DFEAT + kc + c4;
    unsigned dst = (unsigned)(uintptr_t)(lds + row * LDA + c4);  // LDS byte offset
    asm volatile("global_load_async_to_lds_b128 %0, %1, off"
                 :: "v"(dst), "v"(src)
                 : "memory");
  }
}

__device__ __forceinline__ void insert8(float (&nn)[NN], float v) {
  if (v < nn[NN - 1]) {
    nn[NN - 1] = v;
#pragma unroll
    for (int q = NN - 2; q >= 0; --q) {
      float a = nn[q], b = nn[q + 1];
      nn[q] = fminf(a, b);
      nn[q + 1] = fmaxf(a, b);
    }
  }
}

// ---------------- row norms ----------------
__global__ __launch_bounds__(256) void norms_kernel(const float* __restrict__ X,
                                                    float* __restrict__ out) {
  __shared__ float red[256];
  const int row = blockIdx.x;
  const float* x = X + (size_t)row * DFEAT;
  float s = 0.f;
  for (int i = threadIdx.x; i < DFEAT; i += 256) {
    float v = x[i];
    s += v * v;
  }
  red[threadIdx.x] = s;
  __syncthreads();
  for (int st = 128; st > 0; st >>= 1) {
    if (threadIdx.x < st) red[threadIdx.x] += red[threadIdx.x + st];
    __syncthreads();
  }
  if (threadIdx.x == 0) out[row] = red[0];
}

// ---------------- self-distance pass: per-row NN smallest d^2 (partial per gy) ----------------
__global__ __launch_bounds__(256) void radii_kernel(const float* __restrict__ X,
                                                    const float* __restrict__ norms,
                                                    float* __restrict__ partial, int N) {
  __shared__ float As[2][BM * LDA];
  __shared__ float Bs[2][BM * LDA];
  const int tid = threadIdx.x;
  const int wave = tid >> 5;
  const int lane = tid & 31;
  const int lrow = lane & 15;
  const int koff = (lane >> 4) << 1;
  const int r0 = blockIdx.x * BM;
  const int gy = blockIdx.y;
  const int myRow = r0 + wave * 16 + lane;  // meaningful when lane < 16
  const bool rowValid = (lane < 16) && (myRow < N);
  const float nr = rowValid ? norms[myRow] : 0.f;

  float nn[NN];
#pragma unroll
  for (int i = 0; i < NN; ++i) nn[i] = 3.0e38f;

  const int ncc = (N + BN - 1) / BN;
  for (int jc = gy; jc < ncc; jc += GY) {
    const int c0 = jc * BN;
    v8f acc[8] = {};
    __syncthreads();  // scratch-scan readers of As/Bs from previous jc are done
    // software pipeline prologue: async-stage chunk 0 into buffer 0
    stage_async(X, r0, N, 0, &As[0][0], tid);
    stage_async(X, c0, N, 0, &Bs[0][0], tid);
    for (int kc = 0; kc < DFEAT; kc += KC) {
      const int p = (kc / KC) & 1;
      WAIT_ASYNC0();    // my loads for buffer p have landed in LDS
      __syncthreads();  // everyone's landed; everyone done reading buffer p^1
      if (kc + KC < DFEAT) {  // overlap next chunk's DMA with this chunk's WMMAs
        stage_async(X, r0, N, kc + KC, &As[p ^ 1][0], tid);
        stage_async(X, c0, N, kc + KC, &Bs[p ^ 1][0], tid);
      }
#pragma unroll
      for (int kk = 0; kk < KC; kk += 4) {
        v2f a = *(const v2f*)&As[p][(wave * 16 + lrow) * LDA + kk + koff];
#pragma unroll
        for (int t = 0; t < 8; ++t) {
          v2f b = *(const v2f*)&Bs[p][(t * 16 + lrow) * LDA + kk + koff];
          acc[t] = __builtin_amdgcn_wmma_f32_16x16x4_f32(false, a, false, b, (short)0,
                                                         acc[t], false, false);
        }
      }
    }
    // dump dot-products via LDS (reusing staging buffers), scan per-row smallest
    float* scr = ((wave < 4) ? &As[0][0] : &Bs[0][0]) + (wave & 3) * (16 * SCR_LD);
#pragma unroll
    for (int h = 0; h < 2; ++h) {
      __syncthreads();
#pragma unroll
      for (int t = 0; t < 4; ++t) {
#pragma unroll
        for (int r = 0; r < 8; ++r) {
          int lr = r + ((lane >> 4) << 3);
          scr[lr * SCR_LD + t * 16 + (lane & 15)] = acc[h * 4 + t][r];
        }
      }
      __syncthreads();
      if (rowValid) {
        for (int c = 0; c < 64; ++c) {
          int col = c0 + h * 64 + c;
          if (col < N) {
            float d2 = fmaxf(nr + norms[col] - 2.0f * scr[lane * SCR_LD + c], 0.f);
            insert8(nn, d2);
          }
        }
      }
    }
  }
  if (rowValid) {
    float* p = partial + ((size_t)gy * N + myRow) * NN;
#pragma unroll
    for (int i = 0; i < NN; ++i) p[i] = nn[i];
  }
}

// ---------------- merge per-gy partials -> squared radius ----------------
__global__ __launch_bounds__(256) void merge_kernel(const float* __restrict__ partial,
                                                    const int* __restrict__ kptr,
                                                    float* __restrict__ r2, int N) {
  int row = blockIdx.x * blockDim.x + threadIdx.x;
  if (row >= N) return;
  float nn[NN];
#pragma unroll
  for (int i = 0; i < NN; ++i) nn[i] = 3.0e38f;
  for (int g = 0; g < GY; ++g) {
    const float* p = partial + ((size_t)g * N + row) * NN;
#pragma unroll
    for (int i = 0; i < NN; ++i) insert8(nn, p[i]);
  }
  int kk = kptr[0];
  if (kk < 0) kk = 0;
  if (kk > NN - 1) kk = NN - 1;
  r2[row] = nn[kk];  // (k+1)-th smallest squared distance (diagonal 0 included)
}

__global__ void zero_kernel(unsigned* __restrict__ p, int n) {
  int i = blockIdx.x * blockDim.x + threadIdx.x;
  if (i < n) p[i] = 0u;
}

// ---------------- cross pass: fake rows x real cols ----------------
__global__ __launch_bounds__(256) void cross_kernel(
    const float* __restrict__ F, const float* __restrict__ R,
    const float* __restrict__ nF, const float* __restrict__ nR,
    const float* __restrict__ r2F, const float* __restrict__ r2R,
    unsigned* __restrict__ flagF, unsigned* __restrict__ flagR, int N) {
  __shared__ float As[2][BM * LDA];
  __shared__ float Bs[2][BM * LDA];
  __shared__ unsigned char colHit[MAXN];
  const int tid = threadIdx.x;
  const int wave = tid >> 5;
  const int lane = tid & 31;
  const int lrow = lane & 15;
  const int koff = (lane >> 4) << 1;
  const int r0 = blockIdx.x * BM;
  const int gy = blockIdx.y;
  const int myRow = r0 + wave * 16 + lane;
  const bool rowValid = (lane < 16) && (myRow < N);
  const float nr = rowValid ? nF[myRow] : 0.f;
  const float rf = rowValid ? r2F[myRow] : -1.f;
  bool rowHit = false;

  for (int i = tid; i < MAXN; i += 256) colHit[i] = 0;

  const int ncc = (N + BN - 1) / BN;
  for (int jc = gy; jc < ncc; jc += GY) {
    const int c0 = jc * BN;
    v8f acc[8] = {};
    __syncthreads();
    stage_async(F, r0, N, 0, &As[0][0], tid);
    stage_async(R, c0, N, 0, &Bs[0][0], tid);
    for (int kc = 0; kc < DFEAT; kc += KC) {
      const int p = (kc / KC) & 1;
      WAIT_ASYNC0();
      __syncthreads();
      if (kc + KC < DFEAT) {
        stage_async(F, r0, N, kc + KC, &As[p ^ 1][0], tid);
        stage_async(R, c0, N, kc + KC, &Bs[p ^ 1][0], tid);
      }
#pragma unroll
      for (int kk = 0; kk < KC; kk += 4) {
        v2f a = *(const v2f*)&As[p][(wave * 16 + lrow) * LDA + kk + koff];
#pragma unroll
        for (int t = 0; t < 8; ++t) {
          v2f b = *(const v2f*)&Bs[p][(t * 16 + lrow) * LDA + kk + koff];
          acc[t] = __builtin_amdgcn_wmma_f32_16x16x4_f32(false, a, false, b, (short)0,
                                                         acc[t], false, false);
        }
      }
    }
    float* scr = ((wave < 4) ? &As[0][0] : &Bs[0][0]) + (wave & 3) * (16 * SCR_LD);
#pragma unroll
    for (int h = 0; h < 2; ++h) {
      __syncthreads();
#pragma unroll
      for (int t = 0; t < 4; ++t) {
#pragma unroll
        for (int r = 0; r < 8; ++r) {
          int lr = r + ((lane >> 4) << 3);
          scr[lr * SCR_LD + t * 16 + (lane & 15)] = acc[h * 4 + t][r];
        }
      }
      __syncthreads();
      if (rowValid) {
        for (int c = 0; c < 64; ++c) {
          int col = c0 + h * 64 + c;
          if (col < N) {
            float d2 = fmaxf(nr + nR[col] - 2.0f * scr[lane * SCR_LD + c], 0.f);
            if (d2 <= r2R[col]) rowHit = true;  // precision hit for this fake row
            if (d2 <= rf) colHit[col] = 1;      // recall hit for this real col
          }
        }
      }
    }
  }
  if (rowValid && rowHit) flagF[myRow] = 1u;  // idempotent, zero-initialized
  __syncthreads();
  for (int i = tid; i < N; i += 256)
    if (colHit[i]) flagR[i] = 1u;  // idempotent across blocks
}

// ---------------- final means ----------------
__global__ __launch_bounds__(256) void reduce_kernel(const unsigned* __restrict__ flagF,
                                                     const unsigned* __restrict__ flagR,
                                                     float* __restrict__ out, int N) {
  __shared__ float sf[256], sr[256];
  int tid = threadIdx.x;
  float a = 0.f, b = 0.f;
  for (int i = tid; i < N; i += 256) {
    a += flagF[i] ? 1.f : 0.f;
    b += flagR[i] ? 1.f : 0.f;
  }
  sf[tid] = a;
  sr[tid] = b;
  __syncthreads();
  for (int s = 128; s > 0; s >>= 1) {
    if (tid < s) { sf[tid] += sf[tid + s]; sr[tid] += sr[tid + s]; }
    __syncthreads();
  }
  if (tid == 0) {
    out[0] = sf[0] / (float)N;  // precision
    out[1] = sr[0] / (float)N;  // recall
  }
}

extern "C" void kernel_launch(void* const* d_in, const int* in_sizes, int n_in,
                              void* d_out, int out_size, void* d_ws, size_t ws_size,
                              hipStream_t stream) {
  (void)n_in; (void)out_size; (void)ws_size;
  const float* real = (const float*)d_in[0];
  const float* fake = (const float*)d_in[1];
  const int* kptr = (const int*)d_in[2];
  const int N = in_sizes[0] / DFEAT;  // 10000

  float* ws = (float*)d_ws;
  float* normR = ws;                 // [N]
  float* normF = ws + N;             // [N]
  float* r2R = ws + 2 * (size_t)N;   // [N]
  float* r2F = ws + 3 * (size_t)N;   // [N]
  unsigned* flagF = (unsigned*)(ws + 4 * (size_t)N);  // [N]
  unsigned* flagR = flagF + N;                        // [N]
  float* partial = ws + 6 * (size_t)N;                // [GY * N * NN]

  const int nRowBlocks = (N + BM - 1) / BM;
  const dim3 grid2(nRowBlocks, GY);

  norms_kernel<<<N, 256, 0, stream>>>(real, normR);
  norms_kernel<<<N, 256, 0, stream>>>(fake, normF);

  radii_kernel<<<grid2, 256, 0, stream>>>(real, normR, partial, N);
  merge_kernel<<<(N + 255) / 256, 256, 0, stream>>>(partial, kptr, r2R, N);

  radii_kernel<<<grid2, 256, 0, stream>>>(fake, normF, partial, N);
  merge_kernel<<<(N + 255) / 256, 256, 0, stream>>>(partial, kptr, r2F, N);

  zero_kernel<<<(2 * N + 255) / 256, 256, 0, stream>>>(flagF, 2 * N);

  cross_kernel<<<grid2, 256, 0, stream>>>(fake, real, normF, normR, r2F, r2R,
                                          flagF, flagR, N);

  reduce_kernel<<<1, 256, 0, stream>>>(flagF, flagR, (float*)d_out, N);
}